// TensorProduct_54185307406707
// MI455X (gfx1250) — compile-verified
//
#include <hip/hip_runtime.h>
#include <cmath>
#include <cstring>

typedef float v2f __attribute__((ext_vector_type(2)));
typedef float v8f __attribute__((ext_vector_type(8)));
typedef unsigned int uintx4 __attribute__((ext_vector_type(4)));
typedef int intx4 __attribute__((ext_vector_type(4)));
typedef int intx8 __attribute__((ext_vector_type(8)));

#define ZT 16
#define NTHREADS 256
#define XSTR 1156   // LDS row stride for x1 tile (16B aligned, stride%64==4 -> no bank conflicts)

struct W3JTab { float c3[363]; };

// path metadata (compile-time so the 11-path loop fully unrolls)
static constexpr int PI1[11] = {0,0,0,1,1,1,1,2,2,2,2};
static constexpr int PI2[11] = {0,1,2,0,1,1,2,0,1,2,2};
static constexpr int PI3[11] = {0,1,2,1,0,2,1,2,1,0,2};
static constexpr int PC3[11] = {0,1,10,35,44,53,98,143,168,213,238};
static constexpr int LDIM[3] = {1,3,5};
static constexpr int O1v[3]  = {0,128,512};
static constexpr int O2v[3]  = {0,1,4};
static constexpr int ACB[3]  = {0,1,4};   // accumulator base per l3

#if __has_builtin(__builtin_amdgcn_tensor_load_to_lds)
#define HAVE_TDM 1
#else
#define HAVE_TDM 0
#endif

__global__ __launch_bounds__(NTHREADS)
void tp_kernel(const float* __restrict__ x1, const float* __restrict__ x2,
               const float* __restrict__ wt, float* __restrict__ out, W3JTab tab)
{
    __shared__ float x1s[ZT * XSTR];     // x1 tile, padded; reused for output staging
    __shared__ float x2s[ZT * 10];
    __shared__ float As[ZT * 25];        // A[z][k][i], fixed strides 25/5
    __shared__ float Ts[5 * 128 * ZT];   // T[k][u][z]
    __shared__ float c3s[363];

    const int tid  = threadIdx.x;
    const int wave = tid >> 5;
    const int lane = tid & 31;
    const int m    = lane & 15;          // WMMA row (M) / col (N) index
    const int hi   = lane >> 4;          // half-wave selector
    const long z0  = (long)blockIdx.x * ZT;

    for (int i = tid; i < 363; i += NTHREADS) c3s[i] = tab.c3[i];
    for (int i = tid; i < ZT * 9; i += NTHREADS) {
        int z = i / 9, j = i - z * 9;
        x2s[z * 10 + j] = x2[(z0 + z) * 9 + j];
    }

#if HAVE_TDM
    // ---- Tensor Data Mover: DMA the 16x1152 fp32 tile global->LDS, one row per
    // iteration so the LDS destination gets a padded row stride (XSTR dwords).
    if (wave == 0) {
        const unsigned long long ga = (unsigned long long)(uintptr_t)(x1 + z0 * 1152);
        const unsigned ldsa = (unsigned)(uintptr_t)&x1s[0];   // low 32b = LDS byte offset
        uintx4 g0;
        g0[0] = 1u;                                           // count=1 valid descriptor
        g0[1] = ldsa;                                         // lds_addr
        g0[2] = (unsigned)ga;                                 // global_addr[31:0]
        g0[3] = ((unsigned)(ga >> 32) & 0x01FFFFFFu) | (2u << 30); // ga[56:32] | type=2
        intx8 g1;
        g1[0] = (int)((2u << 16) | (1u << 19));               // data_size=4B, iterate_enable
        g1[1] = (int)(1152u << 16);                           // tensor_dim0[15:0]
        g1[2] = (int)(16u << 16);                             // td0[31:16]=0 | tensor_dim1=16
        g1[3] = (int)(1152u << 16);                           // td1[31:16]=0 | tile_dim0=1152
        g1[4] = 1;                                            // tile_dim1=1 | tile_dim2=0
        g1[5] = 1152;                                         // tensor_dim0_stride[31:0]
        g1[6] = (int)(1152u << 16);                           // s0[47:32]=0 | td1_stride[15:0]
        g1[7] = 0;
        intx4 g2;
        g2[0] = 0;                                            // tensor_dim2 (ignored, iterate)
        g2[1] = XSTR;                                         // lds_addr_increment (elems)
        g2[2] = 1152;                                         // global_addr_increment (elems)
        g2[3] = (int)(15u << 16);                             // iterate_count=15 -> 16 rows
        intx4 g3; g3[0] = g3[1] = g3[2] = g3[3] = 0;
#if defined(__clang_major__) && __clang_major__ >= 23
        intx8 gz; for (int q = 0; q < 8; ++q) gz[q] = 0;
        __builtin_amdgcn_tensor_load_to_lds(g0, g1, g2, g3, gz, 0);
#else
        __builtin_amdgcn_tensor_load_to_lds(g0, g1, g2, g3, 0);
#endif
        __builtin_amdgcn_s_wait_tensorcnt(0);
    }
#else
    // fallback: vectorized b128 copy
    for (int q = tid; q < ZT * 288; q += NTHREADS) {
        int z = q / 288, c4 = q - z * 288;
        *reinterpret_cast<float4*>(&x1s[z * XSTR + c4 * 4]) =
            *reinterpret_cast<const float4*>(&x1[(z0 + z) * 1152 + c4 * 4]);
    }
#endif
    __syncthreads();

    v8f acc[9];
    #pragma unroll
    for (int a = 0; a < 9; ++a)
        #pragma unroll
        for (int v = 0; v < 8; ++v) acc[a][v] = 0.0f;

    const int wb = wave << 4;            // this wave's 16-wide w-column block

    #pragma unroll
    for (int n = 0; n < 11; ++n) {
        const int i1 = PI1[n], i2 = PI2[n], i3 = PI3[n];
        const int d1 = LDIM[i1], d2 = LDIM[i2], d3 = LDIM[i3];
        const int o1 = O1v[i1], o2 = O2v[i2], c3o = PC3[n];

        // warm next path's weight slice in L2 (global_prefetch_b8)
        __builtin_prefetch((const void*)(wt + (long)((n + 1) % 11) * 16384 + lane * 64), 0, 1);

        // Phase A: A[z][k][i] = pw * sum_j C3[i,j,k] * x2[z,j]
        for (int e = tid; e < ZT * d3 * d1; e += NTHREADS) {
            int z = e & 15;
            int r = e >> 4;
            int k = r % d3;
            int ii = r / d3;
            float a = 0.0f;
            #pragma unroll
            for (int j = 0; j < d2; ++j)
                a += c3s[c3o + (ii * d2 + j) * d3 + k] * x2s[z * 10 + o2 + j];
            As[z * 25 + k * 5 + ii] = a;
        }
        __syncthreads();

        // Phase T: T[k][u][z] = sum_i A[z][k][i] * x1[z, o1 + u*d1 + i]
        for (int e = tid; e < 2048 * d3; e += NTHREADS) {
            int z = e & 15;
            int u = (e >> 4) & 127;
            int k = e >> 11;
            const float* xp = &x1s[z * XSTR + o1 + u * d1];
            const float* ap = &As[z * 25 + k * 5];
            float s = 0.0f;
            #pragma unroll
            for (int i = 0; i < d1; ++i) s += ap[i] * xp[i];
            Ts[(k * 128 + u) * ZT + z] = s;
        }
        __syncthreads();

        // Phase GEMM: out[z,w,k] += sum_u W[u,w] * T[k,u,z]   (K=128 in 32 steps of 4)
        const long woff = (long)n * 16384;
        for (int c = 0; c < 32; ++c) {
            const int u0 = c * 4 + 2 * hi;
            v2f B;
            B.x = wt[woff + (long)(u0 + 0) * 128 + wb + m];
            B.y = wt[woff + (long)(u0 + 1) * 128 + wb + m];
            #pragma unroll
            for (int k = 0; k < d3; ++k) {
                const float* tp = &Ts[(k * 128 + u0) * ZT + m];
                v2f A;
                A.x = tp[0];
                A.y = tp[ZT];
                acc[ACB[i3] + k] = __builtin_amdgcn_wmma_f32_16x16x4_f32(
                    false, A, false, B, (short)0, acc[ACB[i3] + k], false, false);
            }
        }
        __syncthreads();
    }

    // Restage accumulators into LDS (reuse x1s) for coalesced b128 global stores.
    // C-tile layout: VGPR v, lanes 0-15 -> M=v, lanes 16-31 -> M=v+8; N = lane%16.
    #pragma unroll
    for (int v = 0; v < 8; ++v) {
        const int z = v + 8 * hi;
        x1s[z * XSTR + wb + m] = acc[0][v];
        #pragma unroll
        for (int k = 0; k < 3; ++k)
            x1s[z * XSTR + 128 + (wb + m) * 3 + k] = acc[1 + k][v];
        #pragma unroll
        for (int k = 0; k < 5; ++k)
            x1s[z * XSTR + 512 + (wb + m) * 5 + k] = acc[4 + k][v];
    }
    __syncthreads();
    for (int q = tid; q < ZT * 288; q += NTHREADS) {
        int z = q / 288, c4 = q - z * 288;
        *reinterpret_cast<float4*>(&out[(z0 + z) * 1152 + c4 * 4]) =
            *reinterpret_cast<const float4*>(&x1s[z * XSTR + c4 * 4]);
    }
}

// ---- host: build Wigner-3j tables (e3nn real basis), path weights folded in ----
static void fill_w3j(W3JTab& T) {
    float* t = T.c3;
    std::memset(t, 0, sizeof(T.c3));
    const double pw0 = std::sqrt(1.0 / (3.0 * 128.0));
    const double pw1 = std::sqrt(3.0 / (4.0 * 128.0));
    const double pw2 = std::sqrt(5.0 / (4.0 * 128.0));
    const double c1 = 1.0 / std::sqrt(3.0);
    const double c2 = 1.0 / std::sqrt(5.0);
    const double s  = 1.0 / std::sqrt(10.0);
    const double tt = 2.0 / std::sqrt(30.0);
    const double uu = 1.0 / std::sqrt(30.0);

    t[0] = (float)(1.0 * pw0);                                          // (0,0,0)
    for (int j = 0; j < 3; ++j) t[1  + j * 3 + j] = (float)(c1 * pw1);  // (0,1,1)
    for (int j = 0; j < 5; ++j) t[10 + j * 5 + j] = (float)(c2 * pw2);  // (0,2,2)
    for (int i = 0; i < 3; ++i) t[35 + i * 3 + i] = (float)(c1 * pw1);  // (1,0,1)
    for (int i = 0; i < 3; ++i) t[44 + i * 3 + i] = (float)(c1 * pw0);  // (1,1,0)

    // base (1,1,2) tensor p5[i][j][k], i,j in (y,z,x); k in (xy,yz,z2,zx,x2-y2)
    double p5[45]; std::memset(p5, 0, sizeof(p5));
    auto P5 = [&](int i, int j, int k) -> double& { return p5[i * 15 + j * 5 + k]; };
    P5(2,0,0) = s;  P5(0,2,0) = s;
    P5(0,1,1) = s;  P5(1,0,1) = s;
    P5(1,1,2) = tt; P5(0,0,2) = -uu; P5(2,2,2) = -uu;
    P5(1,2,3) = s;  P5(2,1,3) = s;
    P5(2,2,4) = s;  P5(0,0,4) = -s;
    for (int i = 0; i < 45; ++i) t[53 + i] = (float)(p5[i] * pw2);      // (1,1,2)
    for (int i = 0; i < 3; ++i)                                         // (1,2,1)
        for (int j = 0; j < 5; ++j)
            for (int k = 0; k < 3; ++k)
                t[98 + i * 15 + j * 3 + k] = (float)(p5[i * 15 + k * 5 + j] * pw1);
    for (int i = 0; i < 5; ++i) t[143 + i * 5 + i] = (float)(c2 * pw2); // (2,0,2)
    for (int i = 0; i < 5; ++i)                                         // (2,1,1)
        for (int j = 0; j < 3; ++j)
            for (int k = 0; k < 3; ++k)
                t[168 + i * 9 + j * 3 + k] = (float)(p5[j * 15 + k * 5 + i] * pw1);
    for (int i = 0; i < 5; ++i) t[213 + i * 5 + i] = (float)(c2 * pw0); // (2,2,0)

    // (2,2,2): C[a,b,c] = N * tr(E_a E_b E_c), fully symmetric
    const double N222 = 1.0 / std::sqrt(35.0 / 12.0);
    const double aa = (1.0 / std::sqrt(6.0)) * N222;
    const double bb = (0.5 / std::sqrt(6.0)) * N222;
    const double cc = (0.5 / std::sqrt(2.0)) * N222;
    auto set222 = [&](int x, int y, int z, double v) {
        int p[6][3] = {{x,y,z},{x,z,y},{y,x,z},{y,z,x},{z,x,y},{z,y,x}};
        for (int q = 0; q < 6; ++q)
            t[238 + p[q][0] * 25 + p[q][1] * 5 + p[q][2]] = (float)(v * pw2);
    };
    set222(0,0,2, -aa); set222(1,1,2,  bb); set222(3,3,2, bb);
    set222(1,1,4, -cc); set222(3,3,4,  cc); set222(0,1,3, cc);
    set222(2,2,2,  aa); set222(2,4,4, -aa);
}

extern "C" void kernel_launch(void* const* d_in, const int* in_sizes, int n_in,
                              void* d_out, int out_size, void* d_ws, size_t ws_size,
                              hipStream_t stream) {
    const float* x1 = (const float*)d_in[0];
    const float* x2 = (const float*)d_in[1];
    const float* w  = (const float*)d_in[2];
    float* out = (float*)d_out;
    const int nb = in_sizes[0] / 1152;           // 65536
    W3JTab tab;
    fill_w3j(tab);
    const int grid = nb / ZT;                    // 4096 workgroups
    hipLaunchKernelGGL(tp_kernel, dim3(grid), dim3(NTHREADS), 0, stream,
                       x1, x2, w, out, tab);
}